// EdgeClassify2_5214090297365
// MI455X (gfx1250) — compile-verified
//
#include <hip/hip_runtime.h>

typedef __attribute__((ext_vector_type(2))) float v2f;
typedef __attribute__((ext_vector_type(8))) float v8f;

#define Bn   8
#define An   256
#define Dn   256
#define AKn  320   // A + K
#define NBn  4

// ---------------------------------------------------------------------------
// Kernel 1: h[b] = w1 @ e_output[b] + b1  via V_WMMA_F32_16X16X4_F32.
// One 16x16 output tile per wave, K=320 consumed 4 at a time (80 WMMAs/tile).
// Block = 256 threads = 8 waves; grid = 256 blocks => 2048 tiles (8 batches x
// 16x16 tile grid). EXEC is all-1s (no divergence) as WMMA requires.
// ---------------------------------------------------------------------------
__global__ __launch_bounds__(256) void gemm_h_kernel(
    const float* __restrict__ e,   // (B, 320, 256)
    const float* __restrict__ w1,  // (256, 320)
    const float* __restrict__ b1,  // (256)
    float* __restrict__ h)         // (B, 256, 256)
{
  const int wave = threadIdx.x >> 5;
  const int lane = threadIdx.x & 31;
  const int half = lane >> 4;   // 0: lanes 0-15, 1: lanes 16-31
  const int l    = lane & 15;

  const int tile = blockIdx.x * 8 + wave;   // 0..2047
  const int b    = tile >> 8;               // batch
  const int rem  = tile & 255;
  const int m0   = (rem >> 4) << 4;         // row tile origin (a-dim)
  const int n0   = (rem & 15) << 4;         // col tile origin (d-dim)

  const float* __restrict__ eb   = e + (size_t)b * AKn * Dn;
  const float* __restrict__ arow = w1 + (size_t)(m0 + l) * AKn;

  v8f c = {};
  for (int kk = 0; kk < AKn; kk += 4) {
    const int kb = kk + 2 * half;           // this half-wave's K pair
    v2f a;                                  // A 16x4: lane l holds row m0+l
    a.x = arow[kb];
    a.y = arow[kb + 1];
    v2f bf;                                 // B 4x16: lane l holds col n0+l
    bf.x = eb[(size_t)kb * Dn + n0 + l];
    bf.y = eb[(size_t)(kb + 1) * Dn + n0 + l];
    c = __builtin_amdgcn_wmma_f32_16x16x4_f32(false, a, false, bf,
                                              (short)0, c, false, false);
  }

  // C/D layout: VGPR r -> M = m0 + r + 8*half, N = n0 + l
  float* __restrict__ hb = h + ((size_t)b * An + m0) * Dn + n0 + l;
#pragma unroll
  for (int r = 0; r < 8; ++r) {
    const int m = r + 8 * half;
    hb[(size_t)m * Dn] = c[r] + b1[m0 + m];
  }
}

// ---------------------------------------------------------------------------
// Kernel 2: u[b,a,n] = h[b,a,:] . wf[n, 0:256]
//           v[b,a,n] = h[b,a,:] . wf[n, 256:512]
// 2048 rows x 8 outputs = 16384 threads. Tiny; L2 resident.
// uv layout: u at [0, 2048*4), v at [2048*4, 2048*8) (float4-friendly).
// ---------------------------------------------------------------------------
__global__ __launch_bounds__(256) void proj_uv_kernel(
    const float* __restrict__ h,   // (B*A, 256)
    const float* __restrict__ wf,  // (4, 512)
    float* __restrict__ uv)
{
  const int gid = blockIdx.x * 256 + threadIdx.x;  // 0..16383
  const int row = gid >> 3;                        // b*256 + a
  const int n8  = gid & 7;
  const int n   = n8 & 3;
  const int off = (n8 >> 2) * Dn;                  // 0 => u, 256 => v

  const float* __restrict__ hr = h + (size_t)row * Dn;
  const float* __restrict__ wr = wf + (size_t)n * (2 * Dn) + off;

  float acc = 0.f;
#pragma unroll 4
  for (int d = 0; d < Dn; ++d) acc += hr[d] * wr[d];

  uv[(size_t)(n8 >> 2) * ((size_t)Bn * An * NBn) + (size_t)row * NBn + n] = acc;
}

// ---------------------------------------------------------------------------
// Kernel 3 (bandwidth bound): out[b,i,j,:] = u[b,min(i,j),:] + v[b,max(i,j),:]
//                                            + bf[:]
// One block per (b,i); thread j writes one float4 => fully coalesced 4 KB
// stores per block, 8 MB total (the roofline term).
// ---------------------------------------------------------------------------
__global__ __launch_bounds__(256) void expand_kernel(
    const float* __restrict__ uv,
    const float* __restrict__ bfp,  // (4)
    float* __restrict__ out)        // (B, 256, 256, 4)
{
  const int bi = blockIdx.x;        // b*256 + i
  const int b  = bi >> 8;
  const int i  = bi & 255;
  const int j  = threadIdx.x;
  const int lo = (i < j) ? i : j;
  const int hi = (i < j) ? j : i;

  const float4* __restrict__ u4 = (const float4*)uv;
  const float4* __restrict__ v4 = (const float4*)(uv + (size_t)Bn * An * NBn);

  const float4 uu = u4[b * An + lo];
  const float4 vv = v4[b * An + hi];
  const float4 bb = *(const float4*)bfp;

  float4 o;
  o.x = uu.x + vv.x + bb.x;
  o.y = uu.y + vv.y + bb.y;
  o.z = uu.z + vv.z + bb.z;
  o.w = uu.w + vv.w + bb.w;
  ((float4*)out)[(size_t)bi * An + j] = o;
}

// ---------------------------------------------------------------------------
extern "C" void kernel_launch(void* const* d_in, const int* in_sizes, int n_in,
                              void* d_out, int out_size, void* d_ws, size_t ws_size,
                              hipStream_t stream) {
  const float* e  = (const float*)d_in[0];  // e_output (8, 320, 256)
  const float* w1 = (const float*)d_in[1];  // (256, 320)
  const float* b1 = (const float*)d_in[2];  // (256)
  const float* wf = (const float*)d_in[3];  // (4, 512)
  const float* bf = (const float*)d_in[4];  // (4)
  // d_in[5] = max_atoms (=256), baked into constants.

  float* h  = (float*)d_ws;                       // 2 MB: (8*256, 256) f32
  float* uv = h + (size_t)Bn * An * Dn;           // 64 KB: u then v

  gemm_h_kernel<<<256, 256, 0, stream>>>(e, w1, b1, h);
  proj_uv_kernel<<<64, 256, 0, stream>>>(h, wf, uv);
  expand_kernel<<<Bn * An, 256, 0, stream>>>(uv, bf, (float*)d_out);
}